// _ECELoss_55422257987812
// MI455X (gfx1250) — compile-verified
//
#include <hip/hip_runtime.h>
#include <hip/hip_bf16.h>
#include <math.h>
#include <stdint.h>

#define NUM_BINS 15
#define WAVES_PER_BLOCK 8
#define BLOCK_THREADS (WAVES_PER_BLOCK * 32)
#define MAX_C 256   // row buffer capacity (floats); C=200 here
#define NBUF 3      // async pipeline depth: process r while r+1, r+2 are in flight

struct BinAcc {
    unsigned long long confq[NUM_BINS];  // Q32.32 fixed-point sum of confidences
    unsigned int cnt[NUM_BINS];
    unsigned int acc[NUM_BINS];
};

__global__ void ece_init(BinAcc* ws) {
    int t = threadIdx.x;
    if (t < NUM_BINS) {
        ws->confq[t] = 0ull;
        ws->cnt[t]   = 0u;
        ws->acc[t]   = 0u;
    }
}

__global__ __launch_bounds__(BLOCK_THREADS)
void ece_main(const float* __restrict__ logits, const int* __restrict__ targs,
              int N, int C, BinAcc* __restrict__ ws) {
    __shared__ __align__(16) float rows[WAVES_PER_BLOCK][NBUF][MAX_C];
    __shared__ unsigned long long s_confq[NUM_BINS];
    __shared__ unsigned int s_cnt[NUM_BINS];
    __shared__ unsigned int s_acc[NUM_BINS];

    const int lane = threadIdx.x & 31;
    const int wave = threadIdx.x >> 5;

    if (threadIdx.x < NUM_BINS) {
        s_confq[threadIdx.x] = 0ull;
        s_cnt[threadIdx.x]   = 0u;
        s_acc[threadIdx.x]   = 0u;
    }
    __syncthreads();

    const int wave_global = blockIdx.x * WAVES_PER_BLOCK + wave;
    const int wave_stride = gridDim.x * WAVES_PER_BLOCK;
    const int nchunk = (C + 3) >> 2;            // 16-byte chunks per row (C*4 is a multiple of 16 here)

    // LDS byte offset of this wave's buffer 0 (generic-addr low 32 bits = LDS offset)
    const unsigned lds0 = (unsigned)(uintptr_t)(&rows[wave][0][0]);

    // Issue async global->LDS DMA of one row (ASYNCcnt-tracked, bypasses VGPRs).
    // Data is streamed exactly once -> non-temporal hint keeps it out of L2's way.
    auto issue_row = [&](int row, int which) {
        unsigned long long ga =
            (unsigned long long)(uintptr_t)(logits + (size_t)row * (size_t)C) +
            (unsigned long long)(lane * 16);
        unsigned l = lds0 + (unsigned)(which * (MAX_C * 4)) + (unsigned)(lane * 16);
        if (lane < nchunk) {
            asm volatile("global_load_async_to_lds_b128 %0, %1, off th:TH_LOAD_NT"
                         :: "v"(l), "v"(ga) : "memory");
        }
        if (lane + 32 < nchunk) {
            // INST_OFFSET is added to both the global and LDS addresses.
            asm volatile("global_load_async_to_lds_b128 %0, %1, off offset:512 th:TH_LOAD_NT"
                         :: "v"(l), "v"(ga) : "memory");
        }
    };

    // Per-lane bin accumulators: lane b owns bin b (b < NUM_BINS).
    unsigned int w_cnt = 0u, w_acc = 0u;
    float w_conf = 0.0f;

    // Prime the pipeline: rows r and r+stride in flight.
    if (wave_global < N) issue_row(wave_global, 0);
    if (wave_global + wave_stride < N) issue_row(wave_global + wave_stride, 1);

    int which = 0;
    for (int row = wave_global; row < N; row += wave_stride) {
        const int pre = row + 2 * wave_stride;
        if (pre < N) {
            int wb = which + 2; if (wb >= NBUF) wb -= NBUF;
            issue_row(pre, wb);
            // 6 ops outstanding, in-order completion: <=4 left means current row done.
            asm volatile("s_wait_asynccnt 0x4" ::: "memory");
        } else if (row + wave_stride < N) {
            asm volatile("s_wait_asynccnt 0x2" ::: "memory");
        } else {
            asm volatile("s_wait_asynccnt 0x0" ::: "memory");
        }

        const float* rp = &rows[wave][which][0];

        // Sweep 1: max + first-occurrence argmax (strict > keeps smallest index per lane).
        float m = -3.402823466e+38f;
        int idx = C;
#pragma unroll
        for (int k = 0; k < (MAX_C / 32); ++k) {
            int c = lane + (k << 5);
            if (c < C) {
                float v = rp[c];
                if (v > m) { m = v; idx = c; }
            }
        }
        for (int off = 16; off; off >>= 1) {
            float m2 = __shfl_xor(m, off, 32);
            int   i2 = __shfl_xor(idx, off, 32);
            if (m2 > m || (m2 == m && i2 < idx)) { m = m2; idx = i2; }
        }

        // Sweep 2: softmax denominator with the global row max.
        float s = 0.0f;
#pragma unroll
        for (int k = 0; k < (MAX_C / 32); ++k) {
            int c = lane + (k << 5);
            if (c < C) s += expf(rp[c] - m);
        }
        for (int off = 16; off; off >>= 1) s += __shfl_xor(s, off, 32);

        const float conf = 1.0f / s;           // == max(softmax(row))
        const int targ = targs[row];           // wave-uniform scalar load
        const unsigned int a = (idx == targ) ? 1u : 0u;

        int b = (int)ceilf(conf * (float)NUM_BINS) - 1;
        b = b < 0 ? 0 : (b > NUM_BINS - 1 ? NUM_BINS - 1 : b);

        if (lane == b) { w_cnt += 1u; w_conf += conf; w_acc += a; }

        ++which; if (which >= NBUF) which = 0;
    }

    // Deterministic merge: integer/fixed-point atomics only (order-independent).
    if (lane < NUM_BINS) {
        if (w_cnt) {
            atomicAdd(&s_cnt[lane], w_cnt);
            atomicAdd(&s_acc[lane], w_acc);
            unsigned long long q =
                (unsigned long long)((double)w_conf * 4294967296.0 + 0.5);
            atomicAdd(&s_confq[lane], q);
        }
    }
    __syncthreads();
    if (threadIdx.x < NUM_BINS) {
        int b = threadIdx.x;
        if (s_cnt[b]) {
            atomicAdd(&ws->cnt[b], s_cnt[b]);
            atomicAdd(&ws->acc[b], s_acc[b]);
            atomicAdd(&ws->confq[b], s_confq[b]);
        }
    }
}

__global__ void ece_final(const BinAcc* __restrict__ ws, float* __restrict__ out, int N) {
    if (threadIdx.x == 0 && blockIdx.x == 0) {
        double ece = 0.0;
        for (int b = 0; b < NUM_BINS; ++b) {
            unsigned int cnt = ws->cnt[b];
            if (cnt > 0u) {
                double conf_sum = (double)ws->confq[b] * (1.0 / 4294967296.0);
                double acc_sum  = (double)ws->acc[b];
                double inv      = 1.0 / (double)cnt;
                double gap      = fabs(conf_sum * inv - acc_sum * inv);
                ece += gap * ((double)cnt / (double)N);
            }
        }
        out[0] = (float)ece;
    }
}

extern "C" void kernel_launch(void* const* d_in, const int* in_sizes, int n_in,
                              void* d_out, int out_size, void* d_ws, size_t ws_size,
                              hipStream_t stream) {
    const float* logits = (const float*)d_in[0];
    const int*   targs  = (const int*)d_in[1];
    const int N = in_sizes[1];
    const int C = in_sizes[0] / N;   // 200

    BinAcc* ws = (BinAcc*)d_ws;
    float* out = (float*)d_out;

    hipLaunchKernelGGL(ece_init, dim3(1), dim3(32), 0, stream, ws);

    int blocks = 2048;
    int waves_needed = (N + 31) / 32;  // one wave per 32 rows
    int max_blocks = (waves_needed + WAVES_PER_BLOCK - 1) / WAVES_PER_BLOCK;
    if (blocks > max_blocks) blocks = max_blocks;
    if (blocks < 1) blocks = 1;

    hipLaunchKernelGGL(ece_main, dim3(blocks), dim3(BLOCK_THREADS), 0, stream,
                       logits, targs, N, C, ws);
    hipLaunchKernelGGL(ece_final, dim3(1), dim3(32), 0, stream, ws, out, N);
}